// VectorQuantization_28939489640907
// MI455X (gfx1250) — compile-verified
//
#include <hip/hip_runtime.h>
#include <hip/hip_bf16.h>

// ---------------------------------------------------------------------------
// Vector quantization on MI455X (gfx1250), fp32 path via V_WMMA_F32_16X16X4_F32
//   x:   [32, 128, 64, 64] f32      (B, C, H, W)
//   emb: [512, 128]        f32      (K codes, C channels)
// Outputs (concatenated in d_out, all f32):
//   quantize [32,128,64,64], diff scalar, embedding_ind [32,64,64] (as float)
// ---------------------------------------------------------------------------

typedef __attribute__((ext_vector_type(2))) float v2f;
typedef __attribute__((ext_vector_type(8))) float v8f;

#define BDIM   32
#define CDIM   128
#define HDIM   64
#define WDIM   64
#define KCODES 512
#define CHUNK  64          // codes staged in LDS per iteration
#define NCHUNK (KCODES / CHUNK)
#define QELEMS (BDIM * CDIM * HDIM * WDIM)   // 16777216
#define NBLK   1024        // main-kernel grid

__global__ void __launch_bounds__(256, 2)
vq_main_kernel(const float* __restrict__ x,
               const float* __restrict__ emb,
               float* __restrict__ qout,
               float* __restrict__ indout,
               float* __restrict__ wspart)
{
    // LDS: emb chunk interleaved as [kk(32)][code(64)][cs(4)] floats = 32KB
    __shared__ float s_emb[32 * CHUNK * 4];
    __shared__ float s_esq[CHUNK];
    __shared__ int   s_idx[8][16];
    __shared__ float s_part[8];

    const int tid  = threadIdx.x;
    const int lane = tid & 31;
    const int wv   = tid >> 5;          // 0..7
    const int sub  = lane & 15;         // column / row-in-tile id
    const int half = lane >> 4;         // 0: K=0,1   1: K=2,3 (A & B fragments)

    // wave -> (b, h, w0): waves 0-3 take pair 0, waves 4-7 pair 1
    const int p  = wv >> 2;
    const int w0 = (wv & 3) * 16;
    const int bh = blockIdx.x * 2 + p;
    const int b  = bh >> 6;
    const int h  = bh & 63;

    // base of x[b, 0, h, w0]
    const float* xb = x + (size_t)b * (CDIM * HDIM * WDIM) + (size_t)h * WDIM + w0;

    // -------- load resident A fragments: 16 rows (w) x 128 channels --------
    // A 16x4 f32 layout: lane m=l%16; v0: K = 0/2 (low/high half); v1: K = 1/3
    v2f a[32];
    const int ch0 = half * 2;
#pragma unroll
    for (int kk = 0; kk < 32; ++kk) {
        a[kk].x = xb[(size_t)(4 * kk + ch0)     * (HDIM * WDIM) + sub];
        a[kk].y = xb[(size_t)(4 * kk + ch0 + 1) * (HDIM * WDIM) + sub];
    }

    // running per-lane argmin over (e_sq - 2*dot)
    float smin[8];
    int   simin[8];
#pragma unroll
    for (int r = 0; r < 8; ++r) { smin[r] = 3.0e38f; simin[r] = 0; }

    // -------- sweep code chunks --------
    for (int ci = 0; ci < NCHUNK; ++ci) {
        __syncthreads();   // previous chunk fully consumed
        // stage chunk: global float4 reads are fully linear/coalesced
#pragma unroll
        for (int i = 0; i < 8; ++i) {
            int q = tid + i * 256;                      // 0..2047
            float4 v = *(const float4*)(emb + (size_t)ci * (CHUNK * CDIM) + (size_t)q * 4);
            int code = q >> 5, kk = q & 31;             // q = code*32 + kk
            *(float4*)&s_emb[(kk * CHUNK + code) << 2] = v;
        }
        __syncthreads();
        if (tid < CHUNK) {                              // ||e_k||^2 for this chunk
            float s = 0.0f;
#pragma unroll
            for (int kk = 0; kk < 32; ++kk) {
                float4 v = *(const float4*)&s_emb[(kk * CHUNK + tid) << 2];
                s += v.x * v.x + v.y * v.y + v.z * v.z + v.w * v.w;
            }
            s_esq[tid] = s;
        }
        __syncthreads();

        // 4 column tiles of 16 codes each
        for (int j = 0; j < 4; ++j) {
            const int codeL = j * 16 + sub;
            const float esq = s_esq[codeL];
            v8f acc = {};
#pragma unroll
            for (int kk = 0; kk < 32; ++kk) {
                // B 4x16 f32: lane col=l%16; v0: K=0/2, v1: K=1/3 -> 8B ds read
                v2f bfrag = *(const v2f*)&s_emb[((kk * CHUNK + codeL) << 2) + (half << 1)];
                acc = __builtin_amdgcn_wmma_f32_16x16x4_f32(
                    false, a[kk], false, bfrag, (short)0, acc, false, false);
            }
            const int codeG = ci * CHUNK + j * 16 + sub;
#pragma unroll
            for (int r = 0; r < 8; ++r) {
                float s = esq - 2.0f * acc[r];          // row M = r + 8*half, col codeG
                if (s < smin[r]) { smin[r] = s; simin[r] = codeG; }
            }
        }
    }

    // -------- ||x_row||^2 from resident A registers --------
    float part = 0.0f;
#pragma unroll
    for (int kk = 0; kk < 32; ++kk) part += a[kk].x * a[kk].x + a[kk].y * a[kk].y;
    const float xsq = part + __shfl_xor(part, 16, 32);  // lane l: row l%16

    // -------- cross-lane argmin within each 16-lane group --------
#pragma unroll
    for (int r = 0; r < 8; ++r) {
        float s = smin[r]; int id = simin[r];
#pragma unroll
        for (int off = 8; off >= 1; off >>= 1) {
            float so = __shfl_xor(s, off, 32);
            int   io = __shfl_xor(id, off, 32);
            if (so < s || (so == s && io < id)) { s = so; id = io; }
        }
        smin[r] = s; simin[r] = id;
    }

    // contributor lanes: row r -> lane r (regs r), row 8+r -> lane 24+r (regs r)
    float diffc = 0.0f;
#pragma unroll
    for (int r = 0; r < 8; ++r) {
        bool owner = (lane == r) || (lane == 24 + r);
        if (owner) {
            int m = r + half * 8;                       // row 0..15 in this wave's tile
            s_idx[wv][m] = simin[r];
            int w = w0 + m;
            size_t n = (size_t)b * (WDIM * HDIM) + (size_t)w * HDIM + h;
            indout[n] = (float)simin[r];
            diffc += xsq + smin[r];                     // = ||q - x||^2 for this row
        }
    }
    // wave sum of diff contributions
#pragma unroll
    for (int off = 16; off >= 1; off >>= 1) diffc += __shfl_xor(diffc, off, 32);
    if (lane == 0) s_part[wv] = diffc;

    __syncthreads();

    // -------- gather + write quantize in [B,C,H,W] layout (w-coalesced) --------
    const int myidx = s_idx[wv][sub];
    const float* erow = emb + (size_t)myidx * CDIM;     // L2-resident gather
    float* qb = qout + (size_t)b * (CDIM * HDIM * WDIM) + (size_t)h * WDIM + w0;
#pragma unroll 4
    for (int cp = 0; cp < 64; ++cp) {
        int c = cp * 2 + half;                          // lanes 0-15: even, 16-31: odd
        qb[(size_t)c * (HDIM * WDIM) + sub] = erow[c];
    }

    __syncthreads();
    if (tid == 0) {
        float s = 0.0f;
#pragma unroll
        for (int i = 0; i < 8; ++i) s += s_part[i];
        wspart[blockIdx.x] = s;
    }
}

__global__ void vq_diff_kernel(const float* __restrict__ wspart,
                               float* __restrict__ diff_out)
{
    if (threadIdx.x == 0 && blockIdx.x == 0) {
        float s = 0.0f;
        for (int i = 0; i < NBLK; ++i) s += wspart[i];   // fixed order -> deterministic
        diff_out[0] = 1.25f * s / (float)QELEMS;
    }
}

extern "C" void kernel_launch(void* const* d_in, const int* in_sizes, int n_in,
                              void* d_out, int out_size, void* d_ws, size_t ws_size,
                              hipStream_t stream) {
    const float* x   = (const float*)d_in[0];
    const float* emb = (const float*)d_in[1];
    float* out      = (float*)d_out;
    float* qout     = out;                    // [32,128,64,64]
    float* diff_out = out + QELEMS;           // scalar
    float* ind_out  = out + QELEMS + 1;       // [32,64,64] as float
    float* wspart   = (float*)d_ws;           // 1024 partials (4 KB)

    vq_main_kernel<<<NBLK, 256, 0, stream>>>(x, emb, qout, ind_out, wspart);
    vq_diff_kernel<<<1, 32, 0, stream>>>(wspart, diff_out);
}